// FiveStreamMaha_GCN_Model_74586402063341
// MI455X (gfx1250) — compile-verified
//
#include <hip/hip_runtime.h>

typedef float v2f __attribute__((ext_vector_type(2)));
typedef float v8f __attribute__((ext_vector_type(8)));

#define B_    256
#define T_    150
#define J_    25
#define WAVES 8
#define GSTR  33   // LDS row stride (pad to break bank correlation)

__global__ __launch_bounds__(256)
void knn_feats_kernel(const float* __restrict__ x, float* __restrict__ out) {
    __shared__ float lds_g[WAVES][32 * GSTR];   // per-wave 32x32 Gram tile
    __shared__ float lds_red[WAVES][J_ * 9];    // cross-wave reduction buffer

    const int tid  = threadIdx.x;
    const int wv   = tid >> 5;
    const int lane = tid & 31;
    const int b    = blockIdx.x;

    float acc[9];
#pragma unroll
    for (int f = 0; f < 9; ++f) acc[f] = 0.0f;

    const int  lj      = lane & 15;
    const bool hiHalf  = lane >= 16;
    const int  rowLo   = lj;            // rows 0..15 tile
    const int  rowHi   = 16 + lj;       // rows 16..31 tile (25..31 are pad)
    const bool hiValid = rowHi < J_;

    float* gw = lds_g[wv];

    for (int t = wv; t < T_; t += WAVES) {
        const float* px = x + (size_t)(b * T_ + t) * (J_ * 3);

        // Build f32 16x4 A-operands (K: 0=x,1=y,2=z,3=pad).
        // Lanes 0-15 hold K=0,1 ; lanes 16-31 hold K=2,3 (pad K=3 -> 0).
        v2f aLo, aHi;
        if (!hiHalf) {
            aLo.x = px[rowLo * 3 + 0];
            aLo.y = px[rowLo * 3 + 1];
            aHi.x = hiValid ? px[rowHi * 3 + 0] : 0.0f;
            aHi.y = hiValid ? px[rowHi * 3 + 1] : 0.0f;
        } else {
            aLo.x = px[rowLo * 3 + 2];
            aLo.y = 0.0f;
            aHi.x = hiValid ? px[rowHi * 3 + 2] : 0.0f;
            aHi.y = 0.0f;
        }

        // Gram = P * P^T : for this op the B operand's lane/register contents
        // equal the A operand's, so reuse the same registers.
        v8f cz = {};
        v8f g00 = __builtin_amdgcn_wmma_f32_16x16x4_f32(false, aLo, false, aLo, (short)0, cz, false, false);
        v8f g01 = __builtin_amdgcn_wmma_f32_16x16x4_f32(false, aLo, false, aHi, (short)0, cz, false, false);
        v8f g10 = __builtin_amdgcn_wmma_f32_16x16x4_f32(false, aHi, false, aLo, (short)0, cz, false, false);
        v8f g11 = __builtin_amdgcn_wmma_f32_16x16x4_f32(false, aHi, false, aHi, (short)0, cz, false, false);

        // Spill Gram to LDS. C/D layout: lane<16 -> (M=r, N=lane);
        // lane>=16 -> (M=r+8, N=lane-16).
        const int rbase = hiHalf ? 8 : 0;
#pragma unroll
        for (int r = 0; r < 8; ++r) {
            gw[(rbase + r) * GSTR + lj]            = g00[r];
            gw[(rbase + r) * GSTR + 16 + lj]       = g01[r];
            gw[(16 + rbase + r) * GSTR + lj]       = g10[r];
            gw[(16 + rbase + r) * GSTR + 16 + lj]  = g11[r];
        }
        __builtin_amdgcn_wave_barrier();   // keep DS stores before DS loads

        // Distance matrix is symmetric: lane j scans column j.
        const int j = lane;
        const float nj = gw[j * GSTR + j];        // ||p_j||^2 from diagonal

        float m1 = 3.0e38f, m2 = 3.0e38f, m3 = 3.0e38f, m4 = 3.0e38f;
#pragma unroll 5
        for (int i = 0; i < J_; ++i) {
            float ni  = gw[i * GSTR + i];
            float gij = gw[i * GSTR + j];
            float s   = ni + nj - 2.0f * gij;
            float d   = (s > 0.0f) ? __builtin_sqrtf(s) : 0.0f;
            d = (i == j) ? 3.0e38f : d;           // exclude self (ref drops the 0)
            // branchless sorted-4 insertion
            float t1 = fminf(d,  m1); float r1 = fmaxf(d,  m1);
            float t2 = fminf(r1, m2); float r2 = fmaxf(r1, m2);
            float t3 = fminf(r2, m3); float r3 = fmaxf(r2, m3);
            float t4 = fminf(r3, m4);
            m1 = t1; m2 = t2; m3 = t3; m4 = t4;
        }

        // k=2 / k=3 / k=4 features from sorted m1..m4 (std with ddof=1)
        float mean2 = 0.5f * (m1 + m2);
        float a0 = m1 - mean2, a1 = m2 - mean2;
        float std2 = __builtin_sqrtf(a0 * a0 + a1 * a1);

        float mean3 = (m1 + m2 + m3) * (1.0f / 3.0f);
        float e0 = m1 - mean3, e1 = m2 - mean3, e2 = m3 - mean3;
        float std3 = __builtin_sqrtf((e0 * e0 + e1 * e1 + e2 * e2) * 0.5f);

        float mean4 = (m1 + m2 + m3 + m4) * 0.25f;
        float f0 = m1 - mean4, f1 = m2 - mean4, f2 = m3 - mean4, f3 = m4 - mean4;
        float std4 = __builtin_sqrtf((f0 * f0 + f1 * f1 + f2 * f2 + f3 * f3) * (1.0f / 3.0f));

        if (j < J_) {
            acc[0] += mean2; acc[1] += std2; acc[2] += m1;
            acc[3] += mean3; acc[4] += std3; acc[5] += m1;
            acc[6] += mean4; acc[7] += std4; acc[8] += m1;
        }
        __builtin_amdgcn_wave_barrier();   // next frame's stores stay behind these loads
    }

    // mean over T, then reduce the 8 wave partials
    if (lane < J_) {
#pragma unroll
        for (int f = 0; f < 9; ++f)
            lds_red[wv][lane * 9 + f] = acc[f] * (1.0f / (float)T_);
    }
    __syncthreads();

    if (tid < J_ * 9) {
        float s = 0.0f;
#pragma unroll
        for (int w = 0; w < WAVES; ++w) s += lds_red[w][tid];
        out[(size_t)b * (J_ * 9) + tid] = s;
    }
}

extern "C" void kernel_launch(void* const* d_in, const int* in_sizes, int n_in,
                              void* d_out, int out_size, void* d_ws, size_t ws_size,
                              hipStream_t stream) {
    (void)in_sizes; (void)n_in; (void)d_ws; (void)ws_size; (void)out_size;
    const float* x = (const float*)d_in[0];
    float* out = (float*)d_out;
    knn_feats_kernel<<<B_, 256, 0, stream>>>(x, out);
}